// TDGNN_79628693667938
// MI455X (gfx1250) — compile-verified
//
#include <hip/hip_runtime.h>
#include <hip/hip_bf16.h>
#include <math.h>

// ---- problem constants ----
#define B_      128
#define V_      307
#define T_      12
#define E_      32
#define NPRE_   12
#define VP_     320              // V padded to multiple of 16
#define F3_     96               // 3*E
#define C2E_    64               // 2*E
#define K1_     384              // E*T
#define N1_     768              // 2E*T
#define K2_     19648            // V*2E
#define INROW_  309              // 2+V
#define INB_    4017             // (1+T)*(2+V)
#define WT_     24576            // E*2E*T  (t-stride in dy_weight)
#define WSZ_    294912           // T*E*2E*T (week/date stride in dy_weight)
#define OBSZ_   235776           // V*2E*T  (week/date stride in dy_offset)

typedef __attribute__((ext_vector_type(2))) float v2f;
typedef __attribute__((ext_vector_type(8))) float v8f;

static __device__ __forceinline__ v8f wmma_f32(v2f a, v2f b, v8f c) {
  // V_WMMA_F32_16X16X4_F32 : D = A(16x4) x B(4x16) + C(16x16)
  return __builtin_amdgcn_wmma_f32_16x16x4_f32(false, a, false, b, (short)0, c,
                                               false, false);
}
static __device__ __forceinline__ float leakyf(float x) {
  return x >= 0.f ? x : 0.3f * x;
}
static __device__ __forceinline__ float sigm(float x) {
  return 1.f / (1.f + expf(-x));
}

// ---------- 1) rpe gather ----------
__global__ void k_rpe(const float* __restrict__ srpe,
                      const int* __restrict__ sdist, float* __restrict__ rpe) {
  int i = blockIdx.x * blockDim.x + threadIdx.x;
  if (i < V_ * V_) rpe[i] = srpe[sdist[i]];
}

// ---------- 2) rel = relu(rpe @ rpe^T) ----------
__global__ void k_rel(const float* __restrict__ rpe, float* __restrict__ rel) {
  int i = blockIdx.x * blockDim.x + threadIdx.x;
  if (i >= V_ * V_) return;
  int v = i / V_, g = i % V_;
  const float* a = rpe + (size_t)v * V_;
  const float* b = rpe + (size_t)g * V_;
  float s = 0.f;
  for (int k = 0; k < V_; ++k) s += a[k] * b[k];
  rel[i] = s > 0.f ? s : 0.f;
}

// ---------- 3) rowsum[v] = sum_g rel[v,g] ----------
__global__ void k_rowsum(const float* __restrict__ rel,
                         float* __restrict__ rowsum) {
  int v = blockIdx.x * blockDim.x + threadIdx.x;
  if (v >= V_) return;
  const float* r = rel + (size_t)v * V_;
  float s = 0.f;
  for (int g = 0; g < V_; ++g) s += r[g];
  rowsum[v] = s;
}

// ---------- 4) xr[b,t,v] = sum_g rel[v,g] * x3[b,t,g] ----------
__global__ void k_xr(const float* __restrict__ inp,
                     const float* __restrict__ rel, float* __restrict__ xr) {
  int i = blockIdx.x * blockDim.x + threadIdx.x;
  if (i >= B_ * T_ * V_) return;
  int v = i % V_;
  int bt = i / V_;
  int t = bt % T_, b = bt / T_;
  const float* x3 = inp + (size_t)b * INB_ + (size_t)(1 + t) * INROW_ + 2;
  const float* r = rel + (size_t)v * V_;
  float s = 0.f;
  for (int g = 0; g < V_; ++g) s += r[g] * x3[g];
  xr[i] = s;
}

// ---------- 5) connect: per (t,g) GEMM (128x96)@(96x32), leaky, store
//             h1[b][g][e*T+t]   (WMMA f32 16x16x4) ----------
__global__ void __launch_bounds__(256)
k_connect(const float* __restrict__ inp, const float* __restrict__ xr,
          const float* __restrict__ rowsum, const float* __restrict__ conv_w,
          const float* __restrict__ conv_b, const float* __restrict__ sape,
          const float* __restrict__ dow_emb, const float* __restrict__ tod_emb,
          const float* __restrict__ cw, const float* __restrict__ cb,
          float* __restrict__ h1) {
  __shared__ float xl[B_ * F3_];  // 48 KB
  const int s = blockIdx.x;
  const int t = s / V_, g = s % V_;
  const int tid = threadIdx.x;

  // build xl tile (B x 96) in LDS
  const float rs = rowsum[g];
  for (int idx = tid; idx < B_ * F3_; idx += 256) {
    int b = idx / F3_, f = idx % F3_;
    float val;
    if (f < E_) {
      val = xr[(size_t)(b * T_ + t) * V_ + g] * conv_w[f] + rs * conv_b[f];
    } else if (f < 2 * E_) {
      val = sape[((size_t)g * T_ + t) * E_ + (f - E_)];
    } else {
      const float* row = inp + (size_t)b * INB_ + (size_t)(1 + t) * INROW_;
      int wk = (int)row[0];
      int dy = (int)row[1];
      val = dow_emb[((size_t)wk * V_ + g) * E_ + (f - 2 * E_)] +
            tod_emb[((size_t)dy * V_ + g) * E_ + (f - 2 * E_)];
    }
    xl[idx] = val;
  }
  __syncthreads();

  const int wave = tid >> 5;
  const int l = tid & 31;
  const int l16 = l & 15;
  const int half2 = (l >> 4) * 2;
  const float* Wb = cw + ((size_t)t * V_ + g) * F3_ * E_;  // [f][e] row-major

  v8f acc0 = {0.f, 0.f, 0.f, 0.f, 0.f, 0.f, 0.f, 0.f};
  v8f acc1 = acc0;
  const int m = wave * 16 + l16;  // batch row
  for (int kb = 0; kb < F3_; kb += 4) {
    int k0 = kb + half2;
    v2f a = *(const v2f*)&xl[m * F3_ + k0];
    v2f b0, b1;
    b0.x = Wb[(size_t)k0 * E_ + l16];
    b0.y = Wb[(size_t)(k0 + 1) * E_ + l16];
    b1.x = Wb[(size_t)k0 * E_ + 16 + l16];
    b1.y = Wb[(size_t)(k0 + 1) * E_ + 16 + l16];
    acc0 = wmma_f32(a, b0, acc0);
    acc1 = wmma_f32(a, b1, acc1);
  }
  const float* bias = cb + ((size_t)t * V_ + g) * E_;
  const float bia0 = bias[l16];
  const float bia1 = bias[16 + l16];
#pragma unroll
  for (int r = 0; r < 8; ++r) {
    int bb = wave * 16 + r + 8 * (l >> 4);  // batch index
    size_t base = ((size_t)bb * VP_ + g) * K1_ + t;
    h1[base + (size_t)l16 * T_] = leakyf(acc0[r] + bia0);
    h1[base + (size_t)(16 + l16) * T_] = leakyf(acc1[r] + bia1);
  }
}

// ---------- 6) x_ab: per-b (320x384)@(384x768) twice, combine, leaky,
//             scatter into h2t[b][j][g][i] ----------
__global__ void __launch_bounds__(32)
k_xab(const float* __restrict__ inp, const float* __restrict__ h1,
      const float* __restrict__ dyw1, const float* __restrict__ dyo1,
      const float* __restrict__ dyw2, const float* __restrict__ dyo2,
      float* __restrict__ h2t) {
  const int Nt = blockIdx.x;   // 0..47
  const int Mt = blockIdx.y;   // 0..19
  const int b = blockIdx.z;    // 0..127
  const int l = threadIdx.x;
  const int l16 = l & 15;
  const int half2 = (l >> 4) * 2;

  const int week = (int)inp[(size_t)b * INB_ + 0];
  const int date = (int)inp[(size_t)b * INB_ + 1];
  const float* W1 = dyw1 + (size_t)week * WSZ_;
  const float* W2 = dyw2 + (size_t)date * WSZ_;
  const int n = Nt * 16 + l16;  // 0..767
  const float* Arow = h1 + (size_t)b * VP_ * K1_ + (size_t)(Mt * 16 + l16) * K1_;

  // k % T_ , k / T_ counters for the two gathered B rows (k0, k0+1)
  int t0 = half2, e0 = 0;
  int t1 = half2 + 1, e1 = 0;

  v8f accA = {0.f, 0.f, 0.f, 0.f, 0.f, 0.f, 0.f, 0.f};
  v8f accB = accA;
  for (int kb = 0; kb < K1_; kb += 4) {
    v2f a = *(const v2f*)&Arow[kb + half2];
    v2f b1, b2;
    b1.x = W1[(size_t)t0 * WT_ + (size_t)e0 * N1_ + n];
    b1.y = W1[(size_t)t1 * WT_ + (size_t)e1 * N1_ + n];
    b2.x = W2[(size_t)t0 * WT_ + (size_t)e0 * N1_ + n];
    b2.y = W2[(size_t)t1 * WT_ + (size_t)e1 * N1_ + n];
    accA = wmma_f32(a, b1, accA);
    accB = wmma_f32(a, b2, accB);
    t0 += 4; if (t0 >= T_) { t0 -= T_; ++e0; }
    t1 += 4; if (t1 >= T_) { t1 -= T_; ++e1; }
  }

  const float* bo1 = dyo1 + (size_t)week * OBSZ_;
  const float* bo2 = dyo2 + (size_t)date * OBSZ_;
  const int j = n % T_;      // time index after transpose
  const int i = n / T_;      // channel index (0..63)
#pragma unroll
  for (int r = 0; r < 8; ++r) {
    int g = Mt * 16 + r + 8 * (l >> 4);
    if (g < V_) {
      float bia = 0.5f * (bo1[(size_t)g * N1_ + n] + bo2[(size_t)g * N1_ + n]);
      float v = leakyf(0.5f * (accA[r] + accB[r]) + bia);
      h2t[(((size_t)b * T_ + j) * V_ + g) * C2E_ + i] = v;
    }
  }
}

// ---------- 7) tg: (1536 x 19648) @ (19648 x 307) ----------
__global__ void __launch_bounds__(32)
k_tg(const float* __restrict__ h2t, const float* __restrict__ tge2tg,
     float* __restrict__ tg) {
  const int Nt = blockIdx.x;  // 0..19
  const int Mt = blockIdx.y;  // 0..95
  const int l = threadIdx.x;
  const int l16 = l & 15;
  const int half2 = (l >> 4) * 2;
  const int n = Nt * 16 + l16;
  const bool nv = (n < V_);
  const float* Arow = h2t + (size_t)(Mt * 16 + l16) * K2_;

  v8f acc[4];
#pragma unroll
  for (int u = 0; u < 4; ++u) acc[u] = (v8f){0.f,0.f,0.f,0.f,0.f,0.f,0.f,0.f};

  for (int kb = 0; kb < K2_; kb += 16) {
#pragma unroll
    for (int u = 0; u < 4; ++u) {
      int k0 = kb + u * 4 + half2;
      v2f a = *(const v2f*)&Arow[k0];
      v2f bf = {0.f, 0.f};
      if (nv) {
        bf.x = tge2tg[(size_t)k0 * V_ + n];
        bf.y = tge2tg[(size_t)(k0 + 1) * V_ + n];
      }
      acc[u] = wmma_f32(a, bf, acc[u]);
    }
  }
  v8f accS = acc[0] + acc[1] + acc[2] + acc[3];
  if (nv) {
#pragma unroll
    for (int r = 0; r < 8; ++r) {
      int m = Mt * 16 + r + 8 * (l >> 4);  // = b*T + t
      tg[(size_t)m * V_ + n] = accS[r];
    }
  }
}

// ---------- 8) gt: (B*V x 768) @ (768 x 12), contiguous A chunks ----------
__global__ void k_gt(const float* __restrict__ h2t,
                     const float* __restrict__ gte2gt, float* __restrict__ gt) {
  int i = blockIdx.x * blockDim.x + threadIdx.x;
  if (i >= B_ * V_ * NPRE_) return;
  int o = i % NPRE_;
  int bg = i / NPRE_;
  int g = bg % V_, b = bg / V_;
  const float* a = h2t + (size_t)b * T_ * V_ * C2E_ + (size_t)g * N1_;
  float s = 0.f;
  for (int k = 0; k < N1_; ++k) s += a[k] * gte2gt[(size_t)k * NPRE_ + o];
  gt[i] = s;
}

// ---------- 9) gates + weight1/weight2 + fully + transpose out ----------
__global__ void k_final(const float* __restrict__ inp,
                        const float* __restrict__ tg,
                        const float* __restrict__ gt,
                        const float* __restrict__ w1,
                        const float* __restrict__ w2,
                        const float* __restrict__ fw,
                        const float* __restrict__ fb,
                        float* __restrict__ out) {
  int i = blockIdx.x * blockDim.x + threadIdx.x;
  if (i >= B_ * V_) return;
  int g = i % V_, b = i / V_;

  float g1[T_], g2[T_];
#pragma unroll
  for (int t = 0; t < T_; ++t) {
    float x3 = inp[(size_t)b * INB_ + (size_t)(1 + t) * INROW_ + 2 + g];
    float tv = tg[((size_t)b * T_ + t) * V_ + g];
    float gv = gt[((size_t)b * V_ + g) * NPRE_ + t];  // NPRE_==T_
    g1[t] = (tv + x3) * sigm(tv);
    g2[t] = (gv + x3) * sigm(gv);
  }
  float x1[NPRE_], x2[NPRE_];
#pragma unroll
  for (int o = 0; o < NPRE_; ++o) { x1[o] = 0.f; x2[o] = 0.f; }
  for (int t = 0; t < T_; ++t) {
    const float* r1 = w1 + ((size_t)g * T_ + t) * NPRE_;
    const float* r2 = w2 + ((size_t)g * T_ + t) * NPRE_;
#pragma unroll
    for (int o = 0; o < NPRE_; ++o) {
      x1[o] += r1[o] * g1[t];
      x2[o] += r2[o] * g2[t];
    }
  }
#pragma unroll
  for (int o2 = 0; o2 < NPRE_; ++o2) {
    float s = fb[o2];
    for (int c = 0; c < NPRE_; ++c) {
      s += x1[c] * fw[(size_t)c * NPRE_ + o2];
      s += x2[c] * fw[(size_t)(NPRE_ + c) * NPRE_ + o2];
    }
    out[((size_t)b * NPRE_ + o2) * V_ + g] = s;
  }
}

// ---------------- launch ----------------
extern "C" void kernel_launch(void* const* d_in, const int* in_sizes, int n_in,
                              void* d_out, int out_size, void* d_ws,
                              size_t ws_size, hipStream_t stream) {
  const float* inp   = (const float*)d_in[0];
  const int*   sdist = (const int*)d_in[1];
  const float* conv_w = (const float*)d_in[2];
  const float* conv_b = (const float*)d_in[3];
  const float* srpe   = (const float*)d_in[4];
  const float* sape   = (const float*)d_in[5];
  const float* dow_emb = (const float*)d_in[6];
  const float* tod_emb = (const float*)d_in[7];
  const float* cw     = (const float*)d_in[8];
  const float* cb     = (const float*)d_in[9];
  const float* dyw1   = (const float*)d_in[10];
  const float* dyo1   = (const float*)d_in[11];
  const float* dyw2   = (const float*)d_in[12];
  const float* dyo2   = (const float*)d_in[13];
  const float* tge2tg = (const float*)d_in[14];
  const float* gte2gt = (const float*)d_in[15];
  const float* w1     = (const float*)d_in[16];
  const float* w2     = (const float*)d_in[17];
  const float* fw     = (const float*)d_in[18];
  const float* fb     = (const float*)d_in[19];

  float* ws = (float*)d_ws;
  float* rpe    = ws + 0;          //  94249
  float* rel    = ws + 94464;      //  94249
  float* rowsum = ws + 188928;     //    307
  float* xr     = ws + 189440;     // 471552
  float* h1     = ws + 660992;     // B*VP*K1 = 15728640
  float* h2t    = ws + 16389632;   // B*T*V*2E = 30179328
  float* tg     = ws + 46568960;   // 471552
  float* gt     = ws + 47040512;   // 471552  (end: 47512064 floats ~190MB)

  hipMemsetAsync(h1, 0, (size_t)B_ * VP_ * K1_ * sizeof(float), stream);

  k_rpe<<<(V_ * V_ + 255) / 256, 256, 0, stream>>>(srpe, sdist, rpe);
  k_rel<<<(V_ * V_ + 255) / 256, 256, 0, stream>>>(rpe, rel);
  k_rowsum<<<(V_ + 63) / 64, 64, 0, stream>>>(rel, rowsum);
  k_xr<<<(B_ * T_ * V_ + 255) / 256, 256, 0, stream>>>(inp, rel, xr);
  k_connect<<<T_ * V_, 256, 0, stream>>>(inp, xr, rowsum, conv_w, conv_b, sape,
                                         dow_emb, tod_emb, cw, cb, h1);
  k_xab<<<dim3(N1_ / 16, VP_ / 16, B_), 32, 0, stream>>>(inp, h1, dyw1, dyo1,
                                                         dyw2, dyo2, h2t);
  k_tg<<<dim3(VP_ / 16, (B_ * T_) / 16), 32, 0, stream>>>(h2t, tge2tg, tg);
  k_gt<<<(B_ * V_ * NPRE_ + 255) / 256, 256, 0, stream>>>(h2t, gte2gt, gt);
  k_final<<<(B_ * V_ + 255) / 256, 256, 0, stream>>>(inp, tg, gt, w1, w2, fw,
                                                     fb, (float*)d_out);
}